// MultilevelProposal_20169166422180
// MI455X (gfx1250) — compile-verified
//
#include <hip/hip_runtime.h>
#include <hip/hip_bf16.h>
#include <cstddef>

typedef __attribute__((ext_vector_type(2))) float v2f;
typedef __attribute__((ext_vector_type(8))) float v8f;

#define NPAD 1008          // 63 * 16 tiles (>= max 1000 candidates)
#define NTILE 63
#define WORDS 32           // 32 u32 words = 1024 bits per suppression row
#define PRE_NMS 1000
#define POST_NMS 500
#define NLVL 5
#define BATCH 8
#define NBLK (BATCH * NLVL)
#define NEG_BIG  (-3.0e38f)  // padding sentinel
#define NEG_INF  (-3.4e38f)  // suppression sentinel (< NEG_BIG)
#define BBOX_CLIP 4.135166556742356f
#define NMS_THR 0.7f

__device__ __constant__ int c_N[NLVL] = {196608, 49152, 12288, 3072, 768};

__device__ inline unsigned f2key(float f) {
    unsigned u = __float_as_uint(f);
    return (u & 0x80000000u) ? ~u : (u | 0x80000000u);
}

__device__ inline void decode_clip(const float* dp, const float* ap,
                                   float hm, float wm, float* ob) {
    float ay1 = ap[0], ax1 = ap[1], ay2 = ap[2], ax2 = ap[3];
    float dy = dp[0], dx = dp[1], dh = dp[2], dw = dp[3];
    float ha = ay2 - ay1 + 1.0f, wa = ax2 - ax1 + 1.0f;
    float cya = ay1 + 0.5f * ha, cxa = ax1 + 0.5f * wa;
    dh = fminf(dh, BBOX_CLIP);
    dw = fminf(dw, BBOX_CLIP);
    float cy = dy * ha + cya, cx = dx * wa + cxa;
    float h = __expf(dh) * ha, w = __expf(dw) * wa;
    float y1 = cy - 0.5f * h, x1 = cx - 0.5f * w;
    float y2 = cy + 0.5f * h - 1.0f, x2 = cx + 0.5f * w - 1.0f;
    ob[0] = fminf(fmaxf(y1, 0.0f), hm);
    ob[1] = fminf(fmaxf(x1, 0.0f), wm);
    ob[2] = fminf(fmaxf(y2, 0.0f), hm);
    ob[3] = fminf(fmaxf(x2, 0.0f), wm);
}

// -------------------------------------------------------------------------
// Kernel 1: per-(batch,level) exact top-K via 4-pass radix select on float
// key bits, deterministic chunked compaction, sigmoid + decode + clip, pad
// to NPAD, and zero that block's suppression bit-matrix region.
// -------------------------------------------------------------------------
__global__ __launch_bounds__(256) void k_topk_decode(
    const float* __restrict__ s0, const float* __restrict__ s1,
    const float* __restrict__ s2, const float* __restrict__ s3,
    const float* __restrict__ s4,
    const float* __restrict__ b0, const float* __restrict__ b1,
    const float* __restrict__ b2, const float* __restrict__ b3,
    const float* __restrict__ b4,
    const float* __restrict__ a0, const float* __restrict__ a1,
    const float* __restrict__ a2, const float* __restrict__ a3,
    const float* __restrict__ a4,
    const float* __restrict__ iminfo,
    float* __restrict__ topk_s, float* __restrict__ topk_b,
    unsigned* __restrict__ supp) {
    const int blk = blockIdx.x;
    const int b = blk / NLVL, l = blk % NLVL;
    const int tid = threadIdx.x;
    const int N = c_N[l];
    const int K = (N < PRE_NMS) ? N : PRE_NMS;

    const float *sp, *bp, *ap;
    if (l == 0)      { sp = s0; bp = b0; ap = a0; }
    else if (l == 1) { sp = s1; bp = b1; ap = a1; }
    else if (l == 2) { sp = s2; bp = b2; ap = a2; }
    else if (l == 3) { sp = s3; bp = b3; ap = a3; }
    else             { sp = s4; bp = b4; ap = a4; }
    sp += (size_t)b * N;
    bp += (size_t)b * N * 4;
    ap += (size_t)b * N * 4;

    const float hm = iminfo[b * 5 + 0] - 1.0f;
    const float wm = iminfo[b * 5 + 1] - 1.0f;

    float* os = topk_s + (size_t)blk * NPAD;
    float* ob = topk_b + (size_t)blk * NPAD * 4;

    __shared__ unsigned hist[256];
    __shared__ unsigned sh_d, sh_hi;
    __shared__ unsigned sgt[256], seq[256];

    if (N <= PRE_NMS) {
        // level p6: take everything, in index order
        for (int i = tid; i < N; i += 256) {
            float x = sp[i];
            os[i] = 1.0f / (1.0f + __expf(-x));
            decode_clip(bp + 4 * i, ap + 4 * i, hm, wm, ob + 4 * i);
        }
    } else {
        // --- 4-pass MSB radix select for exact K-th key ---
        unsigned prefix = 0;
        int krem = K;
        for (int pass = 3; pass >= 0; --pass) {
            hist[tid] = 0u;
            __syncthreads();
            const unsigned pmask =
                (pass == 3) ? 0u : (0xFFFFFFFFu << ((pass + 1) * 8));
            const int sh = pass * 8;
            for (int i = tid; i < N; i += 256) {
                unsigned key = f2key(sp[i]);
                if ((key & pmask) == prefix)
                    atomicAdd(&hist[(key >> sh) & 255u], 1u);
            }
            __syncthreads();
            if (tid == 0) {
                unsigned cum = 0, d = 0;
                for (int bin = 255; bin >= 0; --bin) {
                    unsigned c = hist[bin];
                    if (cum + c >= (unsigned)krem) { d = (unsigned)bin; break; }
                    cum += c;
                }
                sh_d = d;
                sh_hi = cum;
            }
            __syncthreads();
            prefix |= (sh_d << sh);
            krem -= (int)sh_hi;
            __syncthreads();
        }
        const unsigned T = prefix;   // exact K-th largest key
        const int ngt = K - krem;    // count strictly greater than T

        // --- deterministic compaction (index-ordered chunks) ---
        const int chunk = (N + 255) / 256;
        const int lo = tid * chunk;
        const int hi = min(N, lo + chunk);
        unsigned cgt = 0, ceq = 0;
        for (int i = lo; i < hi; ++i) {
            unsigned k = f2key(sp[i]);
            cgt += (k > T);
            ceq += (k == T);
        }
        sgt[tid] = cgt;
        seq[tid] = ceq;
        __syncthreads();
        if (tid == 0) {
            unsigned acc = 0;
            for (int t = 0; t < 256; ++t) { unsigned c = sgt[t]; sgt[t] = acc; acc += c; }
            acc = 0;
            for (int t = 0; t < 256; ++t) { unsigned c = seq[t]; seq[t] = acc; acc += c; }
        }
        __syncthreads();
        unsigned ogt = sgt[tid], oeq = seq[tid];
        for (int i = lo; i < hi; ++i) {
            unsigned k = f2key(sp[i]);
            int pos = -1;
            if (k > T) {
                pos = (int)(ogt++);
            } else if (k == T) {
                unsigned r = oeq++;
                if ((int)r < krem) pos = ngt + (int)r;
            }
            if (pos >= 0) {
                float x = sp[i];
                os[pos] = 1.0f / (1.0f + __expf(-x));
                decode_clip(bp + 4 * i, ap + 4 * i, hm, wm, ob + 4 * pos);
            }
        }
    }

    // pad tail to NPAD with -BIG score / zero box
    for (int i = K + tid; i < NPAD; i += 256) {
        os[i] = NEG_BIG;
        float* o = ob + 4 * i;
        o[0] = o[1] = o[2] = o[3] = 0.0f;
    }
    // zero this block's suppression matrix (atomicOr target in kernel 2)
    unsigned* sp2 = supp + (size_t)blk * NPAD * WORDS;
    for (int i = tid; i < NPAD * WORDS; i += 256) sp2[i] = 0u;
}

// -------------------------------------------------------------------------
// Kernel 2: pairwise IoU suppression bit-matrix.
//  - boxes staged into LDS with CDNA5 async global->LDS (ASYNCcnt path)
//  - area_i + area_j per 16x16 tile via one v_wmma_f32_16x16x4_f32
//    (A = [area_m, 1], B = [[1],[area_n]]  ->  D[m][n] = area_m + area_n)
//  - division-free threshold: iou >= thr  <=>  inter >= thr * denom
//  - wave32 __ballot packs 16 column bits per D-VGPR row half
// -------------------------------------------------------------------------
__global__ __launch_bounds__(256) void k_iou_supp(
    const float* __restrict__ topk_b, unsigned* __restrict__ supp) {
    const int blk = blockIdx.x;
    const int tid = threadIdx.x;

    __shared__ float4 bx[NPAD];
    __shared__ float ar[NPAD];

    const float* gb = topk_b + (size_t)blk * NPAD * 4;

    // --- async stage 1008 float4 boxes into LDS ---
    for (int i = tid; i < NPAD; i += 256) {
        unsigned lds_off = (unsigned)(size_t)(&bx[i]);  // low 32 bits = LDS offset
        unsigned goff = (unsigned)(i * 16);
        asm volatile("global_load_async_to_lds_b128 %0, %1, %2"
                     :: "v"(lds_off), "v"(goff), "s"(gb)
                     : "memory");
    }
    asm volatile("s_wait_asynccnt 0" ::: "memory");
    __syncthreads();

    for (int i = tid; i < NPAD; i += 256) {
        float4 v = bx[i];
        ar[i] = fmaxf(v.z - v.x, 0.0f) * fmaxf(v.w - v.y, 0.0f);
    }
    __syncthreads();

    unsigned* sr = supp + (size_t)blk * NPAD * WORDS;
    const int wave = tid >> 5, lane = tid & 31, nwaves = 256 >> 5;
    const bool lo = (lane < 16);
    const int l16 = lane & 15;

    for (int tp = wave; tp < NTILE * NTILE; tp += nwaves) {
        const int ti = tp / NTILE, tj = tp % NTILE;
        const int m0 = ti * 16, n0 = tj * 16;

        // A (16x4 f32): v0 = K0 (area_m) lanes0-15 / K2 (0) lanes16-31
        //               v1 = K1 (1.0)    lanes0-15 / K3 (0) lanes16-31
        v2f A, Bv;
        A[0]  = lo ? ar[m0 + l16] : 0.0f;
        A[1]  = lo ? 1.0f         : 0.0f;
        Bv[0] = lo ? 1.0f         : 0.0f;
        Bv[1] = lo ? ar[n0 + l16] : 0.0f;
        v8f C = {0.f, 0.f, 0.f, 0.f, 0.f, 0.f, 0.f, 0.f};
        v8f D = __builtin_amdgcn_wmma_f32_16x16x4_f32(
            false, A, false, Bv, (short)0, C, false, false);

        const int n = n0 + l16;
        const float4 bn = bx[n];
        const int word = n0 >> 5;
        const int shift = n0 & 31;  // 0 or 16
#pragma unroll
        for (int r = 0; r < 8; ++r) {
            const int m = m0 + r + (lo ? 0 : 8);
            const float4 bm = bx[m];
            float iy1 = fmaxf(bm.x, bn.x), ix1 = fmaxf(bm.y, bn.y);
            float iy2 = fminf(bm.z, bn.z), ix2 = fminf(bm.w, bn.w);
            float inter = fmaxf(iy2 - iy1, 0.0f) * fmaxf(ix2 - ix1, 0.0f);
            float denom = fmaxf(D[r] - inter, 1e-8f);   // area_m + area_n - inter
            // division-free: inter/denom >= thr  <=>  inter >= thr*denom (denom>0)
            bool bit = (inter >= NMS_THR * denom);
            unsigned mask = (unsigned)__ballot(bit);    // wave32: 32 valid bits
            if (lane < 2) {
                // lane 0 -> row m0+r (low 16 ballot bits)
                // lane 1 -> row m0+r+8 (high 16 ballot bits)
                unsigned half = lane ? (mask >> 16) : (mask & 0xFFFFu);
                unsigned w = half << shift;
                if (w) atomicOr(&sr[(size_t)(m0 + r + (lane ? 8 : 0)) * WORDS + word], w);
            }
        }
    }
}

// -------------------------------------------------------------------------
// Kernel 3: greedy NMS (500 steps) per (batch,level) using the bit matrix.
// Matches reference semantics: argmax (tie -> lowest index), record value
// (0 if -inf) and box, suppress row + self.
// -------------------------------------------------------------------------
__global__ __launch_bounds__(256) void k_nms(
    const float* __restrict__ topk_s, const float* __restrict__ topk_b,
    const unsigned* __restrict__ supp,
    float* __restrict__ cand_s, float* __restrict__ cand_b) {
    const int blk = blockIdx.x;
    const int b = blk / NLVL, l = blk % NLVL;
    const int tid = threadIdx.x;

    __shared__ float sc[NPAD];
    __shared__ float rv[256];
    __shared__ int ri[256];
    __shared__ int selIdx;

    const float* gs = topk_s + (size_t)blk * NPAD;
    const float4* gb = (const float4*)(topk_b + (size_t)blk * NPAD * 4);
    const unsigned* sr = supp + (size_t)blk * NPAD * WORDS;

    for (int i = tid; i < NPAD; i += 256) sc[i] = gs[i];
    __syncthreads();

    float* ocs = cand_s + (size_t)(b * NLVL + l) * POST_NMS;
    float* ocb = cand_b + (size_t)(b * NLVL + l) * POST_NMS * 4;

    for (int t = 0; t < POST_NMS; ++t) {
        float v = NEG_INF;
        int mi = 0;
        for (int j = tid; j < NPAD; j += 256) {
            float s = sc[j];
            if (s > v) { v = s; mi = j; }
        }
        rv[tid] = v;
        ri[tid] = mi;
        __syncthreads();
        for (int off = 128; off > 0; off >>= 1) {
            if (tid < off) {
                float v2 = rv[tid + off];
                int i2 = ri[tid + off];
                if (v2 > rv[tid] || (v2 == rv[tid] && i2 < ri[tid])) {
                    rv[tid] = v2;
                    ri[tid] = i2;
                }
            }
            __syncthreads();
        }
        if (tid == 0) {
            int idx = ri[0];
            float s = rv[0];
            bool valid = (s > -1.0e37f);
            ocs[t] = valid ? s : 0.0f;
            float4 bb = valid ? gb[idx] : make_float4(0.f, 0.f, 0.f, 0.f);
            float* o = ocb + t * 4;
            o[0] = bb.x; o[1] = bb.y; o[2] = bb.z; o[3] = bb.w;
            selIdx = idx;
            sc[idx] = NEG_INF;
        }
        __syncthreads();
        const unsigned* row = sr + (size_t)selIdx * WORDS;
        for (int j = tid; j < NPAD; j += 256) {
            if ((row[j >> 5] >> (j & 31)) & 1u) sc[j] = NEG_INF;
        }
        __syncthreads();
    }
}

// -------------------------------------------------------------------------
// Kernel 4: final top-500 of 2500 per batch + box gather.
// Output layout: fs [8,500] then fb [8,500,4], flat.
// -------------------------------------------------------------------------
__global__ __launch_bounds__(256) void k_final_topk(
    const float* __restrict__ cand_s, const float* __restrict__ cand_b,
    float* __restrict__ out) {
    const int b = blockIdx.x;
    const int tid = threadIdx.x;
    const int NC = NLVL * POST_NMS;  // 2500

    __shared__ float sc[NLVL * POST_NMS];
    __shared__ float rv[256];
    __shared__ int ri[256];

    for (int i = tid; i < NC; i += 256) sc[i] = cand_s[(size_t)b * NC + i];
    __syncthreads();

    float* fs = out + (size_t)b * POST_NMS;
    float* fb = out + (size_t)BATCH * POST_NMS + (size_t)b * POST_NMS * 4;

    for (int t = 0; t < POST_NMS; ++t) {
        float v = NEG_INF;
        int mi = 0;
        for (int j = tid; j < NC; j += 256) {
            float s = sc[j];
            if (s > v) { v = s; mi = j; }
        }
        rv[tid] = v;
        ri[tid] = mi;
        __syncthreads();
        for (int off = 128; off > 0; off >>= 1) {
            if (tid < off) {
                float v2 = rv[tid + off];
                int i2 = ri[tid + off];
                if (v2 > rv[tid] || (v2 == rv[tid] && i2 < ri[tid])) {
                    rv[tid] = v2;
                    ri[tid] = i2;
                }
            }
            __syncthreads();
        }
        if (tid == 0) {
            int idx = ri[0];
            fs[t] = rv[0];
            const float* cb = cand_b + ((size_t)b * NC + idx) * 4;
            float* o = fb + t * 4;
            o[0] = cb[0]; o[1] = cb[1]; o[2] = cb[2]; o[3] = cb[3];
            sc[idx] = NEG_INF;
        }
        __syncthreads();
    }
}

extern "C" void kernel_launch(void* const* d_in, const int* in_sizes, int n_in,
                              void* d_out, int out_size, void* d_ws, size_t ws_size,
                              hipStream_t stream) {
    (void)in_sizes; (void)n_in; (void)out_size; (void)ws_size;
    // setup_inputs() dict order: per level i: scores, boxes, anchors; then image_info
    const float* s[NLVL];
    const float* bx[NLVL];
    const float* an[NLVL];
    for (int l = 0; l < NLVL; ++l) {
        s[l]  = (const float*)d_in[3 * l + 0];
        bx[l] = (const float*)d_in[3 * l + 1];
        an[l] = (const float*)d_in[3 * l + 2];
    }
    const float* iminfo = (const float*)d_in[15];

    // workspace layout (floats unless noted):
    float* topk_s = (float*)d_ws;                       // 40*1008
    float* topk_b = topk_s + NBLK * NPAD;               // 40*1008*4
    float* cand_s = topk_b + (size_t)NBLK * NPAD * 4;   // 8*5*500
    float* cand_b = cand_s + BATCH * NLVL * POST_NMS;   // 8*5*500*4
    unsigned* supp = (unsigned*)(cand_b + (size_t)BATCH * NLVL * POST_NMS * 4); // 40*1008*32 u32

    k_topk_decode<<<NBLK, 256, 0, stream>>>(
        s[0], s[1], s[2], s[3], s[4],
        bx[0], bx[1], bx[2], bx[3], bx[4],
        an[0], an[1], an[2], an[3], an[4],
        iminfo, topk_s, topk_b, supp);
    k_iou_supp<<<NBLK, 256, 0, stream>>>(topk_b, supp);
    k_nms<<<NBLK, 256, 0, stream>>>(topk_s, topk_b, supp, cand_s, cand_b);
    k_final_topk<<<BATCH, 256, 0, stream>>>(cand_s, cand_b, (float*)d_out);
}